// SocialEncoder_48352741819114
// MI455X (gfx1250) — compile-verified
//
#include <hip/hip_runtime.h>
#include <stdint.h>

#define EMBED_DIM 64

typedef float v2f __attribute__((ext_vector_type(2)));
typedef float v8f __attribute__((ext_vector_type(8)));

// ---------------------------------------------------------------------------
// helpers
// ---------------------------------------------------------------------------
__device__ __forceinline__ unsigned lds_offset_of(const void* p) {
  // flat address of a __shared__ object: aperture in high bits, LDS byte
  // offset in addr[31:0] (ISA 10.2 aperture translation)
  return (unsigned)(uintptr_t)p;
}

__device__ __forceinline__ void async_load_b32(unsigned lds_off, const void* gaddr) {
  // GLOBAL_LOAD_ASYNC_TO_LDS_B32 (opcode 96): copies a global dword to LDS at vdst, tracked by ASYNCcnt
  asm volatile("global_load_async_to_lds_b32 %0, %1, off"
               :: "v"(lds_off), "v"((unsigned long long)(uintptr_t)gaddr)
               : "memory");
}

__device__ __forceinline__ void atomic_add_f32(float* p, float v) {
  (void)__hip_atomic_fetch_add(p, v, __ATOMIC_RELAXED, __HIP_MEMORY_SCOPE_AGENT);
}

// ---------------------------------------------------------------------------
// 0) zero fill
// ---------------------------------------------------------------------------
__global__ void zero_f32(float* __restrict__ p, long n) {
  long i = (long)blockIdx.x * blockDim.x + threadIdx.x;
  long stride = (long)gridDim.x * blockDim.x;
  for (; i < n; i += stride) p[i] = 0.0f;
}

// ---------------------------------------------------------------------------
// 1) deg[dst] += w   (segment_sum over dst)
// ---------------------------------------------------------------------------
__global__ void deg_kernel(const long long* __restrict__ eidx,  // [2*E], dst at [E..2E)
                           const float* __restrict__ w,
                           float* __restrict__ deg, int E) {
  int e = blockIdx.x * blockDim.x + threadIdx.x;
  if (e < E) {
    int d = (int)eidx[(size_t)E + e];
    atomic_add_f32(&deg[d], w[e]);
  }
}

// 2) deg -> deg_inv_sqrt in place
__global__ void dis_kernel(float* __restrict__ deg, int N) {
  int i = blockIdx.x * blockDim.x + threadIdx.x;
  if (i < N) {
    float d = deg[i];
    deg[i] = (d > 0.0f) ? rsqrtf(d) : 0.0f;
  }
}

// 3) int64 -> int32 indices + norm[e] = dis[src]*w*dis[dst]
__global__ void prep_kernel(const long long* __restrict__ eidx,
                            const float* __restrict__ w,
                            const float* __restrict__ dis,
                            int* __restrict__ src32, int* __restrict__ dst32,
                            float* __restrict__ nrm, int E) {
  int e = blockIdx.x * blockDim.x + threadIdx.x;
  if (e < E) {
    int s = (int)eidx[e];
    int d = (int)eidx[(size_t)E + e];
    src32[e] = s;
    dst32[e] = d;
    nrm[e] = dis[s] * w[e] * dis[d];
  }
}

// ---------------------------------------------------------------------------
// 4) scatter layer: xout[dst] += norm * xin[src]
//    Edge metadata streamed into LDS via async global->LDS loads, double
//    buffered per wave (ASYNCcnt). 32 edges per chunk per wave; 16 lanes /
//    edge, float4 per lane.
// ---------------------------------------------------------------------------
__global__ __launch_bounds__(256) void scatter_kernel(
    const int* __restrict__ src32, const int* __restrict__ dst32,
    const float* __restrict__ nrm,
    const float* __restrict__ xin, float* __restrict__ xout,
    int nchunks) {
  __shared__ unsigned s_src[2][8][32];
  __shared__ unsigned s_dst[2][8][32];
  __shared__ float    s_nrm[2][8][32];

  const int lane = threadIdx.x & 31;
  const int wv   = threadIdx.x >> 5;          // 8 waves / block
  const int gw   = blockIdx.x * 8 + wv;
  const int nw   = gridDim.x * 8;
  const int half = lane >> 4;                 // 0: edge 2t, 1: edge 2t+1
  const int sub  = lane & 15;                 // 4-float column chunk

  int c = gw;
  if (c >= nchunks) return;

  // preload first chunk into buffer 0
  {
    int base = c * 32 + lane;
    async_load_b32(lds_offset_of(&s_src[0][wv][lane]), &src32[base]);
    async_load_b32(lds_offset_of(&s_dst[0][wv][lane]), &dst32[base]);
    async_load_b32(lds_offset_of(&s_nrm[0][wv][lane]), &nrm[base]);
  }

  int buf = 0;
  for (; c < nchunks; c += nw) {
    const int cn = c + nw;
    if (cn < nchunks) {                       // prefetch next chunk
      int base = cn * 32 + lane;
      async_load_b32(lds_offset_of(&s_src[buf ^ 1][wv][lane]), &src32[base]);
      async_load_b32(lds_offset_of(&s_dst[buf ^ 1][wv][lane]), &dst32[base]);
      async_load_b32(lds_offset_of(&s_nrm[buf ^ 1][wv][lane]), &nrm[base]);
      asm volatile("s_wait_asynccnt 0x3" ::: "memory");   // current buf done
    } else {
      asm volatile("s_wait_asynccnt 0x0" ::: "memory");
    }

    #pragma unroll 4
    for (int t = 0; t < 16; ++t) {            // 2 edges per iteration
      const int eId = 2 * t + half;
      const unsigned s = s_src[buf][wv][eId];
      const unsigned d = s_dst[buf][wv][eId];
      const float nmv  = s_nrm[buf][wv][eId];
      const float4 v = *(const float4*)(xin + (size_t)s * EMBED_DIM + sub * 4);
      float* o = xout + (size_t)d * EMBED_DIM + sub * 4;
      atomic_add_f32(o + 0, v.x * nmv);
      atomic_add_f32(o + 1, v.y * nmv);
      atomic_add_f32(o + 2, v.z * nmv);
      atomic_add_f32(o + 3, v.w * nmv);
    }
    buf ^= 1;
  }
}

// tail edges (E not multiple of 32) — plain path
__global__ void scatter_tail(const int* __restrict__ src32, const int* __restrict__ dst32,
                             const float* __restrict__ nrm,
                             const float* __restrict__ xin, float* __restrict__ xout,
                             int e0, int E) {
  int t = blockIdx.x * blockDim.x + threadIdx.x;
  int n = (E - e0) * 16;
  if (t < n) {
    int e = e0 + (t >> 4);
    int col = (t & 15) * 4;
    int s = src32[e], d = dst32[e];
    float nmv = nrm[e];
    const float4 v = *(const float4*)(xin + (size_t)s * EMBED_DIM + col);
    float* o = xout + (size_t)d * EMBED_DIM + col;
    atomic_add_f32(o + 0, v.x * nmv);
    atomic_add_f32(o + 1, v.y * nmv);
    atomic_add_f32(o + 2, v.z * nmv);
    atomic_add_f32(o + 3, v.w * nmv);
  }
}

// ---------------------------------------------------------------------------
// 5) finalize: out = (emb + x1 + x2) * (1/3), done as D = sum_j A_j x B_j
//    with B_j = 16x4 slices of (1/3)*I16, via V_WMMA_F32_16X16X4_F32 (exact
//    f32 math). One wave per 16x16 tile.
//    A layout (ISA 7.12.2, 32-bit A 16x4): lane(0-15)=M, vgpr0=K0|K2,
//    vgpr1=K1|K3 by lane half. B: lane=N, vgpr+half selects K. C/D: vgpr r ->
//    M=r (lanes 0-15) / M=r+8 (lanes 16-31), lane&15 = N.
// ---------------------------------------------------------------------------
__global__ __launch_bounds__(256) void finalize_kernel(
    const float* __restrict__ e0, const float* __restrict__ x1,
    const float* __restrict__ x2, float* __restrict__ out, int N) {
  const int lane = threadIdx.x & 31;
  const int wv = (blockIdx.x * blockDim.x + threadIdx.x) >> 5;
  const int row0 = (wv >> 2) * 16;            // 4 column tiles per row tile
  const int col0 = (wv & 3) * 16;
  if (row0 >= N) return;
  const int half = lane >> 4;
  const int m = lane & 15;
  const float third = 1.0f / 3.0f;

  if (row0 + 16 <= N) {
    const size_t rbase = (size_t)(row0 + m) * EMBED_DIM;
    v8f c = {};
    #pragma unroll
    for (int j = 0; j < 4; ++j) {
      const int k = col0 + 4 * j + 2 * half;  // this lane's two A columns
      v2f a;
      a.x = e0[rbase + k]     + x1[rbase + k]     + x2[rbase + k];
      a.y = e0[rbase + k + 1] + x1[rbase + k + 1] + x2[rbase + k + 1];
      const int r0 = 4 * j + 2 * half;        // K rows this lane's B vgprs hold
      v2f b;
      b.x = (m == r0)     ? third : 0.0f;     // B_j[k][n] = third iff n==4j+k
      b.y = (m == r0 + 1) ? third : 0.0f;
      c = __builtin_amdgcn_wmma_f32_16x16x4_f32(
              /*neg_a=*/false, a, /*neg_b=*/false, b,
              /*c_mod=*/(short)0, c, /*reuse_a=*/false, /*reuse_b=*/false);
    }
    #pragma unroll
    for (int r = 0; r < 8; ++r) {
      const int row = row0 + r + 8 * half;
      out[(size_t)row * EMBED_DIM + col0 + m] = c[r];
    }
  } else {
    const int rem = N - row0;
    for (int idx = lane; idx < rem * 16; idx += 32) {
      const int r = idx >> 4, cc = idx & 15;
      const size_t off = (size_t)(row0 + r) * EMBED_DIM + col0 + cc;
      out[off] = (e0[off] + x1[off] + x2[off]) * third;
    }
  }
}

// ---------------------------------------------------------------------------
// launch
// ---------------------------------------------------------------------------
extern "C" void kernel_launch(void* const* d_in, const int* in_sizes, int n_in,
                              void* d_out, int out_size, void* d_ws, size_t ws_size,
                              hipStream_t stream) {
  const float*     emb  = (const float*)d_in[0];       // [N, 64] f32
  const long long* eidx = (const long long*)d_in[1];   // [2, E] i64
  const float*     w    = (const float*)d_in[2];       // [E] f32
  float* out = (float*)d_out;

  const int N = in_sizes[0] / EMBED_DIM;
  const int E = in_sizes[1] / 2;

  // workspace layout
  int*   src32 = (int*)d_ws;                  // E
  int*   dst32 = src32 + E;                   // E
  float* nrm   = (float*)(dst32 + E);         // E
  float* xA    = nrm + E;                     // N*64
  float* xB    = xA + (size_t)N * EMBED_DIM;  // N*64
  float* deg   = xB + (size_t)N * EMBED_DIM;  // N

  const int TB = 256;

  // degree + normalization
  zero_f32<<<256, TB, 0, stream>>>(deg, N);
  deg_kernel<<<(E + TB - 1) / TB, TB, 0, stream>>>(eidx, w, deg, E);
  dis_kernel<<<(N + TB - 1) / TB, TB, 0, stream>>>(deg, N);
  prep_kernel<<<(E + TB - 1) / TB, TB, 0, stream>>>(eidx, w, deg, src32, dst32, nrm, E);

  const int nchunks = E / 32;
  const int tail0 = nchunks * 32;
  const int scatter_blocks = 1024;            // 8192 waves, ~12 chunks each

  // layer 1: xA = A @ emb
  zero_f32<<<2048, TB, 0, stream>>>(xA, (long)N * EMBED_DIM);
  if (nchunks > 0)
    scatter_kernel<<<scatter_blocks, TB, 0, stream>>>(src32, dst32, nrm, emb, xA, nchunks);
  if (tail0 < E)
    scatter_tail<<<((E - tail0) * 16 + TB - 1) / TB, TB, 0, stream>>>(src32, dst32, nrm, emb, xA, tail0, E);

  // layer 2: xB = A @ xA
  zero_f32<<<2048, TB, 0, stream>>>(xB, (long)N * EMBED_DIM);
  if (nchunks > 0)
    scatter_kernel<<<scatter_blocks, TB, 0, stream>>>(src32, dst32, nrm, xA, xB, nchunks);
  if (tail0 < E)
    scatter_tail<<<((E - tail0) * 16 + TB - 1) / TB, TB, 0, stream>>>(src32, dst32, nrm, xA, xB, tail0, E);

  // finalize: out = (emb + xA + xB) / 3 via WMMA
  const long nthreads = (long)((N + 15) / 16) * 4 * 32;
  finalize_kernel<<<(int)((nthreads + TB - 1) / TB), TB, 0, stream>>>(emb, xA, xB, out, N);
}